// PrefillQAttention_37108517437826
// MI455X (gfx1250) — compile-verified
//
#include <hip/hip_runtime.h>

typedef float v2f __attribute__((ext_vector_type(2)));
typedef float v8f __attribute__((ext_vector_type(8)));
typedef int vint4 __attribute__((vector_size(16)));  // matches async builtin

#if defined(__has_builtin)
#if __has_builtin(__builtin_amdgcn_global_load_async_to_lds_b128) && \
    __has_builtin(__builtin_amdgcn_s_wait_asynccnt)
#define HAVE_ASYNC_LDS 1
#endif
#endif

#define GLOBAL_AS __attribute__((address_space(1)))
#define LDS_AS __attribute__((address_space(3)))

namespace {
constexpr int Bc = 2, Nc = 2048, Hc = 16, Dc = 128, Pc = 7;
constexpr int BHc = Bc * Hc;
constexpr float kScale = 0.08838834764831845f;  // 1/sqrt(128)
constexpr float kNegInf = -3.0e38f;
}

__device__ __forceinline__ v8f wmma_f32_k4(v2f a, v2f b, v8f c) {
  // D = A(16x4,f32) * B(4x16,f32) + C(16x16,f32)
  return __builtin_amdgcn_wmma_f32_16x16x4_f32(false, a, false, b, (short)0, c,
                                               false, false);
}

// 16-byte global -> LDS copy; CDNA5 async path when available.
__device__ __forceinline__ void copy16_g2l(const float* gsrc, float* ldst) {
#if defined(HAVE_ASYNC_LDS)
  __builtin_amdgcn_global_load_async_to_lds_b128(
      (GLOBAL_AS vint4*)(uintptr_t)gsrc, (LDS_AS vint4*)(uintptr_t)ldst, 0, 0);
#else
  *(float4*)ldst = *(const float4*)gsrc;
#endif
}
__device__ __forceinline__ void copy_fence_then_barrier() {
#if defined(HAVE_ASYNC_LDS)
  __builtin_amdgcn_s_wait_asynccnt(0);  // my async writes to LDS landed
#endif
  __syncthreads();                      // everyone's writes visible
}

__device__ __forceinline__ float rmax16(float x) {
  x = fmaxf(x, __shfl_xor(x, 1, 16));
  x = fmaxf(x, __shfl_xor(x, 2, 16));
  x = fmaxf(x, __shfl_xor(x, 4, 16));
  x = fmaxf(x, __shfl_xor(x, 8, 16));
  return x;
}
__device__ __forceinline__ float rsum16(float x) {
  x += __shfl_xor(x, 1, 16);
  x += __shfl_xor(x, 2, 16);
  x += __shfl_xor(x, 4, 16);
  x += __shfl_xor(x, 8, 16);
  return x;
}

// ---------------- 1) LSH hash kernel: one thread per (b,h,n) -----------------
__global__ __launch_bounds__(256) void hash_kernel(
    const float* __restrict__ q, const float* __restrict__ k,
    const float* __restrict__ pd, int* __restrict__ qh, int* __restrict__ kh,
    int* __restrict__ keep) {
  __shared__ float sPd[Dc * Pc];
  const int tid = threadIdx.x;
  for (int e = tid; e < Dc * Pc; e += 256) sPd[e] = pd[e];
  if (blockIdx.x == 0 && tid < 64) keep[tid] = 0;
  __syncthreads();

  const int flat = blockIdx.x * 256 + tid;  // b*H*N + h*N + n  ([B,H,N] layout)
  const int b = flat / (Hc * Nc);
  const int h = (flat / Nc) & (Hc - 1);
  const int n = flat & (Nc - 1);
  const float* qr = q + (((size_t)b * Nc + n) * Hc + h) * Dc;
  const float* kr = k + (((size_t)b * Nc + n) * Hc + h) * Dc;

  float aq[Pc], ak[Pc];
#pragma unroll
  for (int p = 0; p < Pc; ++p) { aq[p] = 0.f; ak[p] = 0.f; }
  for (int d = 0; d < Dc; d += 4) {
    const float4 xq = *(const float4*)(qr + d);
    const float4 xk = *(const float4*)(kr + d);
#pragma unroll
    for (int p = 0; p < Pc; ++p) {
      const float p0 = sPd[(d + 0) * Pc + p];
      const float p1 = sPd[(d + 1) * Pc + p];
      const float p2 = sPd[(d + 2) * Pc + p];
      const float p3 = sPd[(d + 3) * Pc + p];
      aq[p] += xq.x * p0 + xq.y * p1 + xq.z * p2 + xq.w * p3;
      ak[p] += xk.x * p0 + xk.y * p1 + xk.z * p2 + xk.w * p3;
    }
  }
  int bq = 0, bk = 0;
#pragma unroll
  for (int p = 0; p < Pc; ++p) {
    bq |= (aq[p] > 0.f ? 1 : 0) << p;
    bk |= (ak[p] > 0.f ? 1 : 0) << p;
  }
  // unit_hamming permutation == binary-reflected Gray code
  qh[flat] = bq ^ (bq >> 1);
  kh[flat] = bk ^ (bk >> 1);
}

// ---- 2) stable counting sort per (b,h) + keep-flag computation --------------
__global__ __launch_bounds__(128) void sort_kernel(const int* __restrict__ qh,
                                                   const int* __restrict__ kh,
                                                   int* __restrict__ sidx,
                                                   int* __restrict__ keep) {
  __shared__ int sQ[Nc];
  __shared__ int cnt[128];
  __shared__ int base[128];
  const int bh = blockIdx.x;
  const int b = bh >> 4, h = bh & 15;
  const int tid = threadIdx.x;

  for (int n = tid; n < Nc; n += 128) sQ[n] = qh[bh * Nc + n];
  cnt[tid] = 0;
  __syncthreads();
  for (int n = tid; n < Nc; n += 128) atomicAdd(&cnt[sQ[n]], 1);
  __syncthreads();
  if (tid == 0) {
    int s = 0;
    for (int t = 0; t < 128; ++t) { base[t] = s; s += cnt[t]; }
  }
  __syncthreads();

  // thread tid owns hash value tid; scan in n-order -> stable placement
  int pos = base[tid];
  int eq0 = 0, eq1 = 0;
  for (int n = 0; n < Nc; ++n) {
    if (sQ[n] == tid) {
      sidx[bh * Nc + pos] = n;
      if (kh[bh * Nc + pos] == tid) {  // qhash_sorted[pos]==khash[pos]
        if (pos < 1024) eq0 = 1; else eq1 = 1;
      }
      ++pos;
    }
  }
  if (eq0) atomicOr(&keep[b * 32 + h * 2 + 0], 1);
  if (eq1) atomicOr(&keep[b * 32 + h * 2 + 1], 1);
}

// ---- 3) causal flash attention with f32 WMMA --------------------------------
// 8 waves/block; block = 128 sorted queries of one (b,h); shared K/V LDS tiles
// staged with CDNA5 async global->LDS copies.
__global__ __launch_bounds__(256) void attn_kernel(
    const float* __restrict__ q, const float* __restrict__ k,
    const float* __restrict__ v, const int* __restrict__ sidx,
    const int* __restrict__ keep, float* __restrict__ out) {
  constexpr int KP = 132;  // K/V LDS pitch (conflict-free for B-operand reads)
  constexpr int PP = 20;   // P LDS pitch   (conflict-free for A-operand reads)
  __shared__ __align__(16) float Ks[16 * KP];
  __shared__ __align__(16) float Vs[16 * KP];
  __shared__ __align__(16) float Ps[8 * 16 * PP];

  const int blk = blockIdx.x;
  const int qb = blk & 15;   // query block (128 rows) within sequence
  const int bh = blk >> 4;
  const int b = bh >> 4, h = bh & 15;
  const int tid = threadIdx.x;
  const int wave = tid >> 5;
  const int lane = tid & 31;
  const int m = lane & 15;        // row (A) / col (B,C,D) owned by lane
  const int half = lane >> 4;
  const int koff = half << 1;     // K-pair offset inside each 4-chunk

  const int qt_base = qb * 128 + wave * 16;  // this wave's 16 query rows
  const int n_q = sidx[bh * Nc + qt_base + m];
  const float* qptr = q + (((size_t)b * Nc + n_q) * Hc + h) * Dc;

  // Q tile (16x128) pre-gathered into registers in WMMA A layout
  v2f qreg[32];
#pragma unroll
  for (int kk = 0; kk < 32; ++kk) {
    qreg[kk].x = qptr[kk * 4 + koff];
    qreg[kk].y = qptr[kk * 4 + koff + 1];
  }

  float rowmax[8], rowsum[8];
  v8f oacc[8];  // 16x128 output tile: 8 chunks of 16x16 in C layout
#pragma unroll
  for (int r = 0; r < 8; ++r) { rowmax[r] = kNegInf; rowsum[r] = 0.f; }
#pragma unroll
  for (int c = 0; c < 8; ++c)
#pragma unroll
    for (int r = 0; r < 8; ++r) oacc[c][r] = 0.f;

  const int ktmax = qb * 8 + 7;   // last key tile any wave in block needs
  const int mykt = qb * 8 + wave; // last key tile this wave needs (causal)
  float* Pw = Ps + wave * 16 * PP;

  // this thread's two (row,col) slots of the 16x128 tile (float4 granularity)
  const int ldr0 = tid >> 5, ldc0 = (tid & 31) << 2;          // e = tid
  const int ldr1 = (tid + 256) >> 5, ldc1 = ldc0;             // e = tid+256

  for (int kt = 0; kt <= ktmax; ++kt) {
    __syncthreads();  // previous tile fully consumed
    const int kbase = kt * 16;
    // bug-compatible keep index h*4 + n/512; uniform over this 16-row tile.
    // Folding it into the softmax scale reproduces "zeroed K rows score 0"
    // without scaling the staged tile (keeps the copy a pure async DMA).
    const float tscale =
        keep[h * 4 + (kbase >> 9)] ? kScale : 0.0f;
    {
      const size_t g0 = (((size_t)b * Nc + kbase + ldr0) * Hc + h) * Dc + ldc0;
      const size_t g1 = (((size_t)b * Nc + kbase + ldr1) * Hc + h) * Dc + ldc1;
      copy16_g2l(k + g0, &Ks[ldr0 * KP + ldc0]);
      copy16_g2l(k + g1, &Ks[ldr1 * KP + ldc1]);
      copy16_g2l(v + g0, &Vs[ldr0 * KP + ldc0]);
      copy16_g2l(v + g1, &Vs[ldr1 * KP + ldc1]);
    }
    copy_fence_then_barrier();
    if (kt > mykt) continue;  // wave-uniform: EXEC stays all-ones for WMMA

    // S(16x16) = Q(16x128) * K^T(128x16): 32 WMMAs over K-dim,
    // B operands loaded in groups of 8 so several ds loads are in flight.
    v8f s;
#pragma unroll
    for (int r = 0; r < 8; ++r) s[r] = 0.f;
#pragma unroll
    for (int g = 0; g < 4; ++g) {
      v2f bvv[8];
#pragma unroll
      for (int j = 0; j < 8; ++j) {
        const int kk = g * 8 + j;
        bvv[j].x = Ks[m * KP + kk * 4 + koff];
        bvv[j].y = Ks[m * KP + kk * 4 + koff + 1];
      }
#pragma unroll
      for (int j = 0; j < 8; ++j) s = wmma_f32_k4(qreg[g * 8 + j], bvv[j], s);
    }

    // online softmax (rows live across 16-lane halves)
#pragma unroll
    for (int r = 0; r < 8; ++r) {
      const int qi = qt_base + r + 8 * half;
      const int kj = kbase + m;
      float sv = s[r] * tscale;
      if (kj > qi) sv = kNegInf;  // causal in sorted-query index space
      const float tm = rmax16(sv);
      const float nm = fmaxf(rowmax[r], tm);
      const float corr = __expf(rowmax[r] - nm);
      const float p = __expf(sv - nm);
      rowsum[r] = rowsum[r] * corr + rsum16(p);
      rowmax[r] = nm;
#pragma unroll
      for (int c = 0; c < 8; ++c) oacc[c][r] *= corr;
      Pw[(r + 8 * half) * PP + m] = p;  // C layout -> LDS
    }

    // O += P(16x16) * V(16x128): A operands loaded once, reused for 8 chunks
    v2f av[4];
#pragma unroll
    for (int kk = 0; kk < 4; ++kk) {
      av[kk].x = Pw[m * PP + kk * 4 + koff];
      av[kk].y = Pw[m * PP + kk * 4 + koff + 1];
    }
#pragma unroll
    for (int c = 0; c < 8; ++c) {
      v8f acc = oacc[c];
      v2f bvv[4];
#pragma unroll
      for (int kk = 0; kk < 4; ++kk) {
        bvv[kk].x = Vs[(kk * 4 + koff) * KP + c * 16 + m];
        bvv[kk].y = Vs[(kk * 4 + koff + 1) * KP + c * 16 + m];
      }
#pragma unroll
      for (int kk = 0; kk < 4; ++kk) acc = wmma_f32_k4(av[kk], bvv[kk], acc);
      oacc[c] = acc;
    }
  }

  // normalize and scatter back through sort_idx (== unsort via inv_idx)
#pragma unroll
  for (int r = 0; r < 8; ++r) {
    const int qi = qt_base + r + 8 * half;
    const int n_o = sidx[bh * Nc + qi];
    const float inv = 1.f / rowsum[r];
    float* optr = out + (((size_t)b * Nc + n_o) * Hc + h) * Dc;
#pragma unroll
    for (int c = 0; c < 8; ++c) optr[c * 16 + m] = oacc[c][r] * inv;
  }
}

extern "C" void kernel_launch(void* const* d_in, const int* in_sizes, int n_in,
                              void* d_out, int out_size, void* d_ws,
                              size_t ws_size, hipStream_t stream) {
  const float* q = (const float*)d_in[0];
  const float* k = (const float*)d_in[1];
  const float* v = (const float*)d_in[2];
  const float* pd = (const float*)d_in[3];
  float* out = (float*)d_out;

  int* qh = (int*)d_ws;               // [BH*N]
  int* kh = qh + BHc * Nc;            // [BH*N]
  int* sidx = kh + BHc * Nc;          // [BH*N]
  int* keep = sidx + BHc * Nc;        // [64]

  hash_kernel<<<BHc * Nc / 256, 256, 0, stream>>>(q, k, pd, qh, kh, keep);
  sort_kernel<<<BHc, 128, 0, stream>>>(qh, kh, sidx, keep);
  attn_kernel<<<BHc * (Nc / 128), 256, 0, stream>>>(q, k, v, sidx, keep, out);
}